// Model_33217277067318
// MI455X (gfx1250) — compile-verified
//
#include <hip/hip_runtime.h>
#include <hip/hip_bf16.h>

#define D 128
#define KTILE 64

typedef __attribute__((ext_vector_type(2))) float v2f;
typedef __attribute__((ext_vector_type(8))) float v8f;

__device__ __forceinline__ float sigmoidf_(float x) { return 1.0f / (1.0f + __expf(-x)); }

// ---------------------------------------------------------------------------
// Generic WMMA fp32 GEMM:  out = act(A1 @ W1 [+ A2 @ W2] + bias), optional
// per-row masked write (gate[row]==g && level[row]==l).
// Block = 256 threads = 8 wave32; block computes 128 rows x 128 cols.
// Each wave: 16 rows x 128 cols => 8 accumulator tiles of 16x16 (v8f each).
// Weights staged through LDS in 64-row K-tiles (32 KB), stored K-PAIR-PACKED:
//   lds[((k>>1)*128 + col)*2 + (k&1)] = W[k][col]
// so a lane's B fragment (W[kk][col], W[kk+1][col]) is one aligned float2
// -> single ds_load_b64 straight into the even-aligned VGPR pair (no moves).
// A-fragment layout (ISA 7.12.2, 32-bit A 16x4): lanes 0-15 -> K={k,k+1},
// lanes 16-31 -> K={k+2,k+3}, row = lane&15. C/D: VGPR r holds M=r (lanes
// 0-15) and M=r+8 (lanes 16-31), N = lane&15 within each half.
// ---------------------------------------------------------------------------
__global__ __launch_bounds__(256)
void k_gemm(const float* __restrict__ A1, const float* __restrict__ W1,
            const float* __restrict__ A2, const float* __restrict__ W2,
            const float* __restrict__ bias, float* __restrict__ out,
            int act, int g, int l,
            const int* __restrict__ gate, const int* __restrict__ level)
{
    __shared__ float lw[KTILE * D];   // 32 KB weight tile, K-pair packed

    const int tid  = threadIdx.x;
    const int wave = tid >> 5;
    const int lane = tid & 31;
    const int half = lane >> 4;       // 0: lanes 0-15, 1: lanes 16-31
    const int l16  = lane & 15;
    const int rowA = blockIdx.x * 128 + wave * 16 + l16;   // A row for this lane
    const int kOff = half * 2;

    v8f acc[8] = {};

    for (int part = 0; part < 2; ++part) {
        const float* A = part ? A2 : A1;
        const float* W = part ? W2 : W1;
        if (!A) break;
        const float* Arow = A + rowA * D;
        for (int kp = 0; kp < D; kp += KTILE) {
            // prefetch this K-tile's A chunk for our row while waiting on sync
            __builtin_prefetch(Arow + kp, 0, 0);
            __syncthreads();
            // K-pair packed fill: i = k*128 + col
            for (int i = tid; i < KTILE * D; i += 256) {
                const int k = i >> 7, col = i & (D - 1);
                lw[((k >> 1) * D + col) * 2 + (k & 1)] = W[kp * D + i];
            }
            __syncthreads();
            for (int ks = 0; ks < KTILE; ks += 4) {
                const v2f a = *(const v2f*)(Arow + kp + ks + kOff);
                const int pbase = ((ks >> 1) + half) * D;   // (kk>>1)*D
                #pragma unroll
                for (int nt = 0; nt < 8; ++nt) {
                    const v2f b = *(const v2f*)(lw + (pbase + nt * 16 + l16) * 2);
                    acc[nt] = __builtin_amdgcn_wmma_f32_16x16x4_f32(
                        false, a, false, b, (short)0, acc[nt], false, false);
                }
            }
        }
    }

    // Store: lane holds column nt*16 + l16, rows rowS..rowS+7.
    const int rowS = blockIdx.x * 128 + wave * 16 + half * 8;
    #pragma unroll
    for (int nt = 0; nt < 8; ++nt) {
        const int col = nt * 16 + l16;
        const float bv = bias ? bias[col] : 0.0f;
        #pragma unroll
        for (int r = 0; r < 8; ++r) {
            const int row = rowS + r;
            if (g >= 0 && !(gate[row] == g && level[row] == l)) continue;
            float v = acc[nt][r] + bv;
            if (act == 1)      v = fmaxf(v, 0.0f);
            else if (act == 2) v = tanhf(v);
            out[row * D + col] = v;
        }
    }
}

// ---------------------------------------------------------------------------
// Helper kernels
// ---------------------------------------------------------------------------
__global__ __launch_bounds__(256)
void k_zero(float* __restrict__ p, int n)
{
    int i = blockIdx.x * 256 + threadIdx.x;
    if (i < n) p[i] = 0.0f;
}

__global__ __launch_bounds__(256)
void k_deg(const int* __restrict__ dst, float* __restrict__ deg, int E)
{
    int e = blockIdx.x * 256 + threadIdx.x;
    if (e < E) atomicAdd(&deg[dst[e]], 1.0f);
}

__global__ __launch_bounds__(256)
void k_deg_fin(float* __restrict__ deg_io, float* __restrict__ dis, int n)
{
    int i = blockIdx.x * 256 + threadIdx.x;
    if (i < n) {
        float dv = deg_io[i] + 1.0f;
        dis[i]    = rsqrtf(dv);
        deg_io[i] = 1.0f / dv;   // becomes invdeg
    }
}

// accum[dst] += coef * feat[src]; coef = dis[s]*dis[d] (GCN) or 1; optional
// (gate,level) predicate on dst.  One thread per (edge, feature).
__global__ __launch_bounds__(256)
void k_scatter(const float* __restrict__ feat, float* __restrict__ accum,
               const int* __restrict__ src, const int* __restrict__ dstI,
               const float* __restrict__ dis,
               const int* __restrict__ gate, const int* __restrict__ level,
               int g, int l, int E)
{
    int idx = blockIdx.x * 256 + threadIdx.x;
    if (idx >= E * D) return;
    int e = idx >> 7, f = idx & (D - 1);
    int d = dstI[e];
    if (g >= 0 && !(gate[d] == g && level[d] == l)) return;
    int s = src[e];
    float c = dis ? dis[s] * dis[d] : 1.0f;
    atomicAdd(&accum[d * D + f], c * feat[s * D + f]);
}

// out = scat + invdeg[row]*x + bias[col]   (GCN combine)
__global__ __launch_bounds__(256)
void k_gcn_combine(const float* __restrict__ scat, const float* __restrict__ x,
                   const float* __restrict__ invdeg, const float* __restrict__ b,
                   float* __restrict__ out, int n)
{
    int idx = blockIdx.x * 256 + threadIdx.x;
    if (idx >= n) return;
    int row = idx >> 7, col = idx & (D - 1);
    out[idx] = scat[idx] + invdeg[row] * x[idx] + b[col];
}

// zero hf rows where gate[row]==g && level[row]==l (before masked scatter-set)
__global__ __launch_bounds__(256)
void k_mask_zero(float* __restrict__ hf, const int* __restrict__ gate,
                 const int* __restrict__ level, int g, int l, int n)
{
    int idx = blockIdx.x * 256 + threadIdx.x;
    if (idx >= n) return;
    int row = idx >> 7;
    if (gate[row] == g && level[row] == l) hf[idx] = 0.0f;
}

// Decoder: one wave32 per edge. Lane loads float4 of zs[src], zt[dst],
// wave-reduce dot with shfl_xor, then fused sigmoid-weighted scatter-add.
__global__ __launch_bounds__(256)
void k_decoder(const float* __restrict__ zs, const float* __restrict__ zt,
               const int* __restrict__ src, const int* __restrict__ dstI,
               float* __restrict__ hf, int E)
{
    int gtid = blockIdx.x * 256 + threadIdx.x;
    int e = gtid >> 5;
    int lane = threadIdx.x & 31;
    if (e >= E) return;
    int s = src[e], d = dstI[e];
    const float4 a = *(const float4*)(zs + s * D + lane * 4);
    const float4 b = *(const float4*)(zt + d * D + lane * 4);
    float dot = a.x * b.x + a.y * b.y + a.z * b.z + a.w * b.w;
    #pragma unroll
    for (int off = 16; off > 0; off >>= 1) dot += __shfl_xor(dot, off, 32);
    float w = sigmoidf_(dot);
    float* o = hf + d * D + lane * 4;
    atomicAdd(o + 0, w * a.x);
    atomicAdd(o + 1, w * a.y);
    atomicAdd(o + 2, w * a.z);
    atomicAdd(o + 3, w * a.w);
}

// ---------------------------------------------------------------------------
extern "C" void kernel_launch(void* const* d_in, const int* in_sizes, int n_in,
                              void* d_out, int out_size, void* d_ws, size_t ws_size,
                              hipStream_t stream)
{
    (void)n_in; (void)out_size; (void)ws_size;

    const float* hs_init = (const float*)d_in[0];
    const int*   edge    = (const int*)d_in[1];
    const int*   gate    = (const int*)d_in[2];
    const int*   level   = (const int*)d_in[3];
    const float* gcn_w = (const float*)d_in[4],  *gcn_b = (const float*)d_in[5];
    const float* mu_w  = (const float*)d_in[6],  *mu_b  = (const float*)d_in[7];
    /* d_in[8]=ls_w, d_in[9]=ls_b: eval mode, logstd unused */
    const float* as_w  = (const float*)d_in[10], *as_b  = (const float*)d_in[11];
    const float* ns_w  = (const float*)d_in[12], *ns_b  = (const float*)d_in[13];
    const float* af_w  = (const float*)d_in[14], *af_b  = (const float*)d_in[15];
    const float* nf_w  = (const float*)d_in[16], *nf_b  = (const float*)d_in[17];
    const float* ua_w  = (const float*)d_in[18], *ua_b  = (const float*)d_in[19];
    const float* un_w  = (const float*)d_in[20], *un_b  = (const float*)d_in[21];
    const float* dec_ws= (const float*)d_in[22], *dec_wt= (const float*)d_in[23];

    const int N = in_sizes[0] / D;
    const int E = in_sizes[1] / 2;
    const int* srcI = edge;        // edge_index[0]
    const int* dstI = edge + E;    // edge_index[1]

    float* hs = (float*)d_out;                       // [N,D]
    float* hf = (float*)d_out + (size_t)N * D;       // [N,D]

    float* W0     = (float*)d_ws;                    // x (GCN pre-scatter)
    float* W1     = W0 + (size_t)N * D;              // msg / fmsg / zs
    float* W2     = W1 + (size_t)N * D;              // agg / scatter / zt
    float* dis    = W2 + (size_t)N * D;              // rsqrt(deg+1)
    float* invdeg = dis + N;                         // raw deg -> 1/(deg+1)

    const int ND = N * D, ED = E * D;
    dim3 T(256);
    auto cdiv = [](int a, int b) { return (a + b - 1) / b; };
    const int gND = cdiv(ND, 256), gED = cdiv(ED, 256);
    const int gE  = cdiv(E, 256),  gN  = cdiv(N, 256);
    const int gG  = N / 128,       gDec = cdiv(E * 32, 256);

    // --- degree / normalization ---
    k_zero   <<<gN, T, 0, stream>>>(invdeg, N);
    k_deg    <<<gE, T, 0, stream>>>(dstI, invdeg, E);
    k_deg_fin<<<gN, T, 0, stream>>>(invdeg, dis, N);
    // hf starts at zero
    k_zero   <<<gND, T, 0, stream>>>(hf, ND);

    // --- GCN encoder ---
    k_gemm<<<gG, T, 0, stream>>>(hs_init, gcn_w, nullptr, nullptr, nullptr, W0,
                                 0, -1, 0, nullptr, nullptr);
    k_zero   <<<gND, T, 0, stream>>>(W2, ND);
    k_scatter<<<gED, T, 0, stream>>>(W0, W2, srcI, dstI, dis,
                                     nullptr, nullptr, -1, 0, E);
    k_gcn_combine<<<gND, T, 0, stream>>>(W2, W0, invdeg, gcn_b, W1, ND);

    // --- VAE head (eval): hs = mu ---
    k_gemm<<<gG, T, 0, stream>>>(W1, mu_w, nullptr, nullptr, mu_b, hs,
                                 0, -1, 0, nullptr, nullptr);

    // --- level-wise propagation ---
    for (int l = 1; l <= 8; ++l) {
        // AND gates
        k_gemm<<<gG, T, 0, stream>>>(hs, as_w, nullptr, nullptr, as_b, W1,
                                     1, -1, 0, nullptr, nullptr);          // msg
        k_zero   <<<gND, T, 0, stream>>>(W2, ND);
        k_scatter<<<gED, T, 0, stream>>>(W1, W2, srcI, dstI, nullptr,
                                         gate, level, 1, l, E);            // agg
        k_gemm<<<gG, T, 0, stream>>>(hs, ua_w, W2, ua_w + D * D, ua_b, hs,
                                     1, 1, l, gate, level);                // hs upd
        k_gemm<<<gG, T, 0, stream>>>(hs, af_w, hf, af_w + D * D, af_b, W1,
                                     1, -1, 0, nullptr, nullptr);          // fmsg
        k_mask_zero<<<gND, T, 0, stream>>>(hf, gate, level, 1, l, ND);
        k_scatter  <<<gED, T, 0, stream>>>(W1, hf, srcI, dstI, nullptr,
                                           gate, level, 1, l, E);          // hf upd
        // NOT gates
        k_gemm<<<gG, T, 0, stream>>>(hs, ns_w, nullptr, nullptr, ns_b, W1,
                                     1, -1, 0, nullptr, nullptr);          // msg
        k_zero   <<<gND, T, 0, stream>>>(W2, ND);
        k_scatter<<<gED, T, 0, stream>>>(W1, W2, srcI, dstI, nullptr,
                                         gate, level, 2, l, E);            // agg
        k_gemm<<<gG, T, 0, stream>>>(W2, un_w, nullptr, nullptr, un_b, hs,
                                     2, 2, l, gate, level);                // hs upd
        k_gemm<<<gG, T, 0, stream>>>(hf, nf_w, nullptr, nullptr, nf_b, W1,
                                     1, -1, 0, nullptr, nullptr);          // fmsg
        k_mask_zero<<<gND, T, 0, stream>>>(hf, gate, level, 2, l, ND);
        k_scatter  <<<gED, T, 0, stream>>>(W1, hf, srcI, dstI, nullptr,
                                           gate, level, 2, l, E);          // hf upd
    }

    // --- decoder ---
    k_gemm<<<gG, T, 0, stream>>>(hs, dec_ws, nullptr, nullptr, nullptr, W1,
                                 0, -1, 0, nullptr, nullptr);              // zs
    k_gemm<<<gG, T, 0, stream>>>(hs, dec_wt, nullptr, nullptr, nullptr, W2,
                                 0, -1, 0, nullptr, nullptr);              // zt
    k_zero   <<<gND, T, 0, stream>>>(hf, ND);
    k_decoder<<<gDec, T, 0, stream>>>(W1, W2, srcI, dstI, hf, E);
}